// Conv2dLocal_7086696038450
// MI455X (gfx1250) — compile-verified
//
#include <hip/hip_runtime.h>

// Per-pixel depthwise 3x3 local conv (GuideConv), fp32.
//   input  [B, C, H, W]         = [2, 32, 256, 512]
//   weight [B, C*9, H, W]
//   out    [B, C, H, W]
// Memory-bound (AI ~0.41 FLOP/B): stream weight once with NT b128 loads,
// stage input tile (+halo) in LDS via gfx1250 async global->LDS b128 loads.

#define KB_  2
#define KC_  32
#define KH_  256
#define KW_  512
#define TH   8
#define TW   128
#define LDS_STRIDE 136        // floats per LDS row: 544 B = 34*16 B -> b128 aligned
#define NTHREADS   256        // 8 wave32 waves

typedef float f4 __attribute__((ext_vector_type(4)));

__global__ __launch_bounds__(NTHREADS) void guideconv_kernel(
    const float* __restrict__ input,
    const float* __restrict__ weight,
    float* __restrict__ out)
{
    // LDS row layout (indices 0..135): [0..2 pad][3: left halo][4..131: interior]
    //                                  [132: right halo][133..135 pad]
    // Global column gw maps to LDS index (gw - w0 + 4).
    __shared__ float tile[(TH + 2) * LDS_STRIDE];

    const int bx    = blockIdx.x;
    const int plane = bx >> 7;          // 128 tiles per (b,c) plane
    const int rem   = bx & 127;
    const int ty    = rem >> 2;         // 0..31  -> h tile
    const int tx    = rem & 3;          // 0..3   -> w tile
    const int h0    = ty * TH;
    const int w0    = tx * TW;
    const int tid   = threadIdx.x;

    const size_t HW       = (size_t)KH_ * KW_;
    const float* in_plane = input + (size_t)plane * HW;

    // ---- stage interior: (TH+2) rows x 32 float4 via async b128 loads ----
    for (int idx = tid; idx < (TH + 2) * (TW / 4); idx += NTHREADS) {
        const int r  = idx >> 5;        // 0..9
        const int k  = idx & 31;        // float4 column within interior
        const int gh = h0 + r - 1;
        float* lp = &tile[r * LDS_STRIDE + 4 + k * 4];
        if ((unsigned)gh < (unsigned)KH_) {
            const unsigned lds_addr =
                (unsigned)(unsigned long long)(void*)lp;   // low 32b = LDS offset
            const unsigned long long gaddr =
                (unsigned long long)(const void*)(in_plane + (size_t)gh * KW_ + w0 + k * 4);
            asm volatile("global_load_async_to_lds_b128 %0, %1, off"
                         :
                         : "v"(lds_addr), "v"(gaddr)
                         : "memory");
        } else {
            *(f4*)lp = (f4)0.0f;        // top/bottom halo rows outside image
        }
    }

    // ---- stage left/right halo columns: (TH+2) rows x 2 scalars ----
    if (tid < 2 * (TH + 2)) {
        const int r    = tid >> 1;      // 0..9
        const int side = tid & 1;       // 0 = left, 1 = right
        const int gh   = h0 + r - 1;
        const int gw   = side ? (w0 + TW) : (w0 - 1);
        const int L    = side ? (TW + 4) : 3;
        float* lp = &tile[r * LDS_STRIDE + L];
        if ((unsigned)gh < (unsigned)KH_ && (unsigned)gw < (unsigned)KW_) {
            const unsigned lds_addr =
                (unsigned)(unsigned long long)(void*)lp;
            const unsigned long long gaddr =
                (unsigned long long)(const void*)(in_plane + (size_t)gh * KW_ + gw);
            asm volatile("global_load_async_to_lds_b32 %0, %1, off"
                         :
                         : "v"(lds_addr), "v"(gaddr)
                         : "memory");
        } else {
            *lp = 0.0f;
        }
    }

    asm volatile("s_wait_asynccnt 0x0" ::: "memory");
    __syncthreads();

    // ---- compute: 4 consecutive output pixels per thread ----
    const int row = tid >> 5;          // 0..7 within tile
    const int wq  = (tid & 31) * 4;    // 0,4,...,124 within tile

    // 3x6 input window (taps j=0..2 for 4 pixels) from LDS
    float x[3][6];
#pragma unroll
    for (int i = 0; i < 3; ++i)
#pragma unroll
        for (int c = 0; c < 6; ++c)
            x[i][c] = tile[(row + i) * LDS_STRIDE + wq + 3 + c];

    const int gh = h0 + row;
    const f4* wptr = (const f4*)(weight + (size_t)plane * 9 * HW
                                        + (size_t)gh * KW_ + w0 + wq);
    const size_t tap4 = HW / 4;        // tap-plane stride in float4 units

    f4 acc = (f4)0.0f;
#pragma unroll
    for (int i = 0; i < 3; ++i) {
#pragma unroll
        for (int j = 0; j < 3; ++j) {
            const int t = i * 3 + j;
            const f4 wv = __builtin_nontemporal_load(wptr + (size_t)t * tap4);
            acc.x = __builtin_fmaf(x[i][j + 0], wv.x, acc.x);
            acc.y = __builtin_fmaf(x[i][j + 1], wv.y, acc.y);
            acc.z = __builtin_fmaf(x[i][j + 2], wv.z, acc.z);
            acc.w = __builtin_fmaf(x[i][j + 3], wv.w, acc.w);
        }
    }

    f4* op = (f4*)(out + (size_t)plane * HW + (size_t)gh * KW_ + w0 + wq);
    __builtin_nontemporal_store(acc, op);
}

extern "C" void kernel_launch(void* const* d_in, const int* in_sizes, int n_in,
                              void* d_out, int out_size, void* d_ws, size_t ws_size,
                              hipStream_t stream) {
    (void)in_sizes; (void)n_in; (void)out_size; (void)d_ws; (void)ws_size;
    const float* input  = (const float*)d_in[0];
    const float* weight = (const float*)d_in[1];
    float*       out    = (float*)d_out;

    const int tiles = KB_ * KC_ * (KH_ / TH) * (KW_ / TW);  // 8192
    guideconv_kernel<<<dim3(tiles), dim3(NTHREADS), 0, stream>>>(input, weight, out);
}